// SparseEncoder4D_83348135346670
// MI455X (gfx1250) — compile-verified
//
#include <hip/hip_runtime.h>
#include <math.h>

// ---------------------------------------------------------------------------
// Scene constants (match reference)
// ---------------------------------------------------------------------------
#define XD 256
#define YD 256
#define ZD 32
#define TD 5
#define VTOT (TD * ZD * YD * XD)   // 10,485,760 voxels
#define BMWORDS (VTOT / 32)        // 327,680 u32 occupancy bitmask
#define COUT 8
#define KOFF 81                    // 3^4 kernel offsets
#define KPAD 84                    // padded K for 4-wide WMMA steps (21 steps)

typedef __attribute__((ext_vector_type(2))) float v2f;
typedef __attribute__((ext_vector_type(8))) float v8f;

// ---------------------------------------------------------------------------
// Kernel 1: zero the metadata region (counter + bitmask)
// ---------------------------------------------------------------------------
__global__ void zero_meta_kernel(unsigned* __restrict__ ws, int nwords) {
  int i = blockIdx.x * blockDim.x + threadIdx.x;
  if (i < nwords) ws[i] = 0u;
}

// ---------------------------------------------------------------------------
// Kernel 2: zero the dense output volume (HBM-store bound; b128 stores)
// ---------------------------------------------------------------------------
__global__ void zero_out_kernel(float* __restrict__ out, long long n) {
  long long i = (long long)blockIdx.x * blockDim.x + threadIdx.x;
  long long stride = (long long)gridDim.x * blockDim.x;
  long long n4 = n >> 2;
  float4* o4 = reinterpret_cast<float4*>(out);
  float4 z; z.x = 0.f; z.y = 0.f; z.z = 0.f; z.w = 0.f;
  for (long long j = i; j < n4; j += stride) o4[j] = z;
  for (long long j = (n4 << 2) + i; j < n; j += stride) out[j] = 0.f;
}

// ---------------------------------------------------------------------------
// Kernel 3: point -> voxel quantization, set occupancy bit
// ---------------------------------------------------------------------------
__global__ void scatter_kernel(const float* __restrict__ pts,
                               unsigned* __restrict__ bm, int n) {
  int i = blockIdx.x * blockDim.x + threadIdx.x;
  if (i >= n) return;
  float px = pts[4 * i + 0];
  float py = pts[4 * i + 1];
  float pz = pts[4 * i + 2];
  float pt = pts[4 * i + 3];
  int cx = (int)floorf((px - (-25.6f)) / 0.2f);
  int cy = (int)floorf((py - (-25.6f)) / 0.2f);
  int cz = (int)floorf((pz - (-3.2f)) / 0.2f);
  int ct = (int)floorf(pt);
  cx = min(max(cx, 0), XD - 1);
  cy = min(max(cy, 0), YD - 1);
  cz = min(max(cz, 0), ZD - 1);
  ct = min(max(ct, 0), TD - 1);
  unsigned vid = (((unsigned)ct * ZD + (unsigned)cz) * YD + (unsigned)cy) * XD + (unsigned)cx;
  atomicOr(&bm[vid >> 5], 1u << (vid & 31u));
}

// ---------------------------------------------------------------------------
// Kernel 4: compact occupied voxel ids (wave-aggregated atomics, wave32)
// ---------------------------------------------------------------------------
__global__ void compact_kernel(const unsigned* __restrict__ bm,
                               unsigned* __restrict__ counter,
                               unsigned* __restrict__ vids) {
  unsigned v = blockIdx.x * blockDim.x + threadIdx.x;
  bool act = false;
  if (v < (unsigned)VTOT) act = ((bm[v >> 5] >> (v & 31u)) & 1u) != 0u;
#if __has_builtin(__builtin_amdgcn_ballot_w32)
  unsigned bal = __builtin_amdgcn_ballot_w32(act);
#else
  unsigned bal = (unsigned)__ballot(act);
#endif
  unsigned lane = threadIdx.x & 31u;
  unsigned pre = __builtin_popcount(bal & ((lane == 0u) ? 0u : ((1u << lane) - 1u)));
  unsigned tot = __builtin_popcount(bal);
  unsigned wbase = 0u;
  if (lane == 0u && tot != 0u) wbase = atomicAdd(counter, tot);
  wbase = __builtin_amdgcn_readfirstlane(wbase);  // all lanes converged here
  if (act) vids[wbase + pre] = v;
}

// ---------------------------------------------------------------------------
// Kernel 5: per-occupied-voxel sparse conv as a 16x16x84 mask-GEMM on
// V_WMMA_F32_16X16X4_F32. One wave per 16-voxel tile; 4 waves/block.
//
// Phase A (branchless, hoisted): each lane packs its 42 neighbor-occupancy
// probes into two 32-bit registers (mlo: steps 0..15, mhi: steps 16..20).
// Validity = per-dim precomputed 12-bit mask indexed by LDS table digits;
// no divides, no divergent exec around loads.
// Phase B (fully unrolled): 21 x V_WMMA_F32_16X16X4_F32 with A bits from
// registers (32-bit tests -> cndmask) and B pairs from LDS Wsum.
// EXEC stays all-ones throughout.
//
// Layouts (ISA 7.12.2): A 16x4 f32 -> lane L: row M=L&15, VGPR0 = K{0|2},
// VGPR1 = K{1|3} by lane half; B mirrors the K mapping with N=L&15 across
// lanes. C/D: lane L, VGPR r -> M = r + 8*(L>=16), N = L&15.
// ---------------------------------------------------------------------------
__global__ __launch_bounds__(128) void conv_kernel(const float* __restrict__ W,
                                                   const unsigned* __restrict__ bm,
                                                   const unsigned* __restrict__ counter,
                                                   const unsigned* __restrict__ vids,
                                                   float* __restrict__ out) {
  __shared__ float wsum[KPAD][16];  // Wsum[k][f] = sum_c W[k][c][f]; zero-padded
  __shared__ int2 offtab[KPAD];     // {flat nid delta, packed digits | okbit<<8}

  int tid = threadIdx.x;  // 0..127, 4 waves
  for (int i = tid; i < KPAD * 16; i += 128) {
    int k = i >> 4, f = i & 15;
    float s = 0.f;
    if (k < KOFF && f < COUT) {
      #pragma unroll
      for (int c = 0; c < 8; ++c) s += W[(k * 8 + c) * 8 + f];
    }
    wsum[k][f] = s;
  }
  if (tid < KPAD) {
    int k = tid;
    int dx = 0, dy = 0, dz = 0, dt = 0, okbit = 0;
    if (k < KOFF) {
      dx = k / 27 - 1;
      dy = (k / 9) % 3 - 1;
      dz = (k / 3) % 3 - 1;
      dt = k % 3 - 1;
      okbit = 1;
    }
    int delta = dx + XD * dy + XD * YD * dz + XD * YD * ZD * dt;
    int dig = (dx + 1) | ((dy + 1) << 2) | ((dz + 1) << 4) | ((dt + 1) << 6) | (okbit << 8);
    offtab[k] = make_int2(delta, dig);
  }
  __syncthreads();

  unsigned cnt = *counter;
  int wave = tid >> 5;
  int lane = tid & 31;
  unsigned base = (blockIdx.x * 4u + (unsigned)wave) * 16u;
  if (base >= cnt) return;  // uniform per wave; EXEC stays all-ones below

  int row = lane & 15;            // M for A, N for B
  int khalf = (lane >> 4) * 2;    // K sub-pair selected by lane half
  unsigned ridx = base + (unsigned)row;
  bool rowValid = ridx < cnt;
  unsigned vid = rowValid ? vids[ridx] : 0u;
  int x = (int)(vid & (XD - 1));
  int y = (int)((vid >> 8) & (YD - 1));
  int z = (int)((vid >> 16) & (ZD - 1));
  int t = (int)(vid >> 21);

  // 12-bit per-dimension validity mask: bits [d*3 + (off+1)] for off in {-1,0,1}
  unsigned vmask = 0u;
  if (rowValid) {
    vmask = ((x > 0) ? 1u : 0u) | 2u | ((x < XD - 1) ? 4u : 0u)
          | (((y > 0) ? 1u : 0u) | 2u | ((y < YD - 1) ? 4u : 0u)) << 3
          | (((z > 0) ? 1u : 0u) | 2u | ((z < ZD - 1) ? 4u : 0u)) << 6
          | (((t > 0) ? 1u : 0u) | 2u | ((t < TD - 1) ? 4u : 0u)) << 9;
  }

  // Phase A: pack 42 branchless occupancy probes into two 32-bit words.
  // Step s, element j -> bit (2s+j): s<16 in mlo, s>=16 in mhi (bit 2s+j-32).
  unsigned mlo = 0u, mhi = 0u;
  #pragma unroll
  for (int s = 0; s < 21; ++s) {
    #pragma unroll
    for (int j = 0; j < 2; ++j) {
      int k = 4 * s + khalf + j;
      int2 od = offtab[k];
      int dg = od.y;
      unsigned ok = (vmask >> (dg & 3)) &
                    (vmask >> (3 + ((dg >> 2) & 3))) &
                    (vmask >> (6 + ((dg >> 4) & 3))) &
                    (vmask >> (9 + ((dg >> 6) & 3))) &
                    (unsigned)(dg >> 8) & 1u;
      unsigned nid = (unsigned)((int)vid + od.x);
      unsigned nid_s = ok ? nid : 0u;              // clamp index, keep load legal
      unsigned bit = (bm[nid_s >> 5] >> (nid_s & 31u)) & ok;
      if (s < 16) mlo |= bit << (2 * s + j);
      else        mhi |= bit << (2 * s + j - 32);
    }
  }

  // Phase B: 21 unrolled WMMA steps, EXEC all-ones, no branches, 32-bit tests
  v8f acc = {};
  #pragma unroll
  for (int s = 0; s < 21; ++s) {
    unsigned word = (s < 16) ? mlo : mhi;
    int sh = (s < 16) ? (2 * s) : (2 * s - 32);
    v2f A, B;
    A.x = ((word >> sh) & 1u) ? 1.0f : 0.0f;
    A.y = ((word >> (sh + 1)) & 1u) ? 1.0f : 0.0f;
    B.x = wsum[4 * s + khalf][row];
    B.y = wsum[4 * s + khalf + 1][row];
    acc = __builtin_amdgcn_wmma_f32_16x16x4_f32(
        /*neg_a=*/false, A, /*neg_b=*/false, B,
        /*c_mod=*/(short)0, acc, /*reuse_a=*/false, /*reuse_b=*/false);
  }

  // ReLU + scattered store: lane's column = output channel (only 0..7 real)
  int ncol = lane & 15;
  if (ncol < COUT) {
    int mhiM = (lane >> 4) * 8;
    #pragma unroll
    for (int r = 0; r < 8; ++r) {
      unsigned ri = base + (unsigned)(mhiM + r);
      if (ri < cnt) {
        unsigned vv = vids[ri];
        float val = acc[r];
        out[(size_t)ncol * (size_t)VTOT + vv] = val > 0.f ? val : 0.f;
      }
    }
  }
}

// ---------------------------------------------------------------------------
// Host-side launcher
// ---------------------------------------------------------------------------
extern "C" void kernel_launch(void* const* d_in, const int* in_sizes, int n_in,
                              void* d_out, int out_size, void* d_ws, size_t ws_size,
                              hipStream_t stream) {
  const float* pts = (const float*)d_in[0];  // [1, N, 4] xyzt
  const float* W   = (const float*)d_in[1];  // [81, 8, 8]
  float* out = (float*)d_out;                // [1, 8, 5, 32, 256, 256]

  unsigned* ws      = (unsigned*)d_ws;
  unsigned* counter = ws;                 // 1 u32 (padded to 16 for alignment)
  unsigned* bm      = ws + 16;            // BMWORDS u32 occupancy bitmask
  unsigned* vids    = ws + 16 + BMWORDS;  // up to N compacted voxel ids

  int n = in_sizes[0] / 4;

  int metaWords = 16 + BMWORDS;
  zero_meta_kernel<<<(metaWords + 255) / 256, 256, 0, stream>>>(ws, metaWords);
  zero_out_kernel<<<4096, 256, 0, stream>>>(out, (long long)out_size);
  scatter_kernel<<<(n + 255) / 256, 256, 0, stream>>>(pts, bm, n);
  compact_kernel<<<(VTOT + 255) / 256, 256, 0, stream>>>(bm, counter, vids);

  int tiles = (n + 15) / 16;          // <= ceil(N/16) occupied-voxel tiles
  int blocks = (tiles + 3) / 4;       // 4 waves (tiles) per block
  conv_kernel<<<blocks, 128, 0, stream>>>(W, bm, counter, vids, out);
}